// HardKNNMask_27762668601762
// MI455X (gfx1250) — compile-verified
//
#include <hip/hip_runtime.h>
#include <math.h>

// Problem constants (from reference): x[4096,256], x_n[16384,256], top-33.
#define QROWS   4096
#define NCOLS   16384
#define KD      256
#define TOPK    33
#define NCHUNK  32          // B columns staged per LDS chunk (2 x 16-col tiles)
#define ASTRIDE 260         // 256 + 4 : TDM pad(4 dw / 256 dw) -> conflict-free A reads
#define BSTRIDE 257         // 256 + 1 : TDM pad(1 dw / 256 dw) -> conflict-free B reads

typedef float v2f __attribute__((ext_vector_type(2)));
typedef float v8f __attribute__((ext_vector_type(8)));
typedef unsigned int u32x4 __attribute__((ext_vector_type(4)));
typedef int i32x4 __attribute__((ext_vector_type(4)));
typedef int i32x8 __attribute__((ext_vector_type(8)));

// ---------------------------------------------------------------------------
// Tensor Data Mover: 2D tile (tile_d1 rows x tile_d0 elems, fp32) from global
// to LDS, with optional LDS row padding (CDNA5 ISA "8.").
// pad_interval_code: DWORDs between pads = 2<<code ; pad_amount_code: pad = code+1 DW.
// ---------------------------------------------------------------------------
__device__ static inline unsigned lds_off(const void* p) {
  // Generic LDS VA = {shared aperture, offset}; low 32 bits are the LDS offset.
  return (unsigned)(unsigned long long)p;
}

__device__ static inline void tdm_load_2d(unsigned lds_addr, const void* gptr,
                                          unsigned tile_d0, unsigned tile_d1,
                                          unsigned tensor_d0, unsigned tensor_d1,
                                          unsigned stride_d0,
                                          unsigned pad_interval_code,
                                          unsigned pad_amount_code) {
  unsigned long long ga = (unsigned long long)gptr;
  u32x4 g0;
  g0[0] = 1u;                                       // count=1, user descriptor
  g0[1] = lds_addr;                                 // lds_addr (bytes)
  g0[2] = (unsigned)(ga & 0xffffffffull);           // global_addr[31:0]
  g0[3] = (unsigned)((ga >> 32) & 0x01ffffffull)    // global_addr[56:32]
          | (2u << 30);                             // type = 2 ("image")
  i32x8 g1;
  g1[0] = (int)((2u << 16)                          // data_size = 4 bytes
          | (1u << 20)                              // pad_enable
          | (pad_interval_code << 22)
          | (pad_amount_code << 25));
  g1[1] = (int)((tensor_d0 & 0xffffu) << 16);       // tensor_dim0[15:0]
  g1[2] = (int)(((tensor_d0 >> 16) & 0xffffu) | ((tensor_d1 & 0xffffu) << 16));
  g1[3] = (int)(((tensor_d1 >> 16) & 0xffffu) | ((tile_d0 & 0xffffu) << 16));
  g1[4] = (int)(tile_d1 & 0xffffu);                 // tile_dim1 ; tile_dim2 = 0
  g1[5] = (int)stride_d0;                           // tensor_dim0_stride[31:0]
  g1[6] = 0;                                        // stride hi / dim1_stride = 0
  g1[7] = 0;
  i32x4 z4 = {0, 0, 0, 0};
  i32x8 z8 = {0, 0, 0, 0, 0, 0, 0, 0};
  // clang-23 / therock form: 6 args (g0, g1, g2, g3, extra, cpol).
  __builtin_amdgcn_tensor_load_to_lds(g0, g1, z4, z4, z8, 0);
}

// ---------------------------------------------------------------------------
// Kernel 1: fill the 256 MB output with -inf (the write-bound floor: ~11 us
// at 23.3 TB/s). Everything not in the top-33 stays -inf.
// ---------------------------------------------------------------------------
__global__ __launch_bounds__(256) void fill_neg_inf(float4* __restrict__ out,
                                                    long n4) {
  const float ninf = -__builtin_huge_valf();
  float4 v = make_float4(ninf, ninf, ninf, ninf);
  long i = (long)blockIdx.x * blockDim.x + threadIdx.x;
  long stride = (long)gridDim.x * blockDim.x;
  for (; i < n4; i += stride) out[i] = v;
}

// ---------------------------------------------------------------------------
// Kernel 2: row squared-norms, one wave32 per row, shfl_xor reduction.
// ---------------------------------------------------------------------------
__global__ __launch_bounds__(256) void row_norms(const float* __restrict__ x,
                                                 const float* __restrict__ xn,
                                                 float* __restrict__ ws) {
  int wid  = (blockIdx.x * blockDim.x + threadIdx.x) >> 5;
  int lane = threadIdx.x & 31;
  if (wid >= QROWS + NCOLS) return;
  const float* src = (wid < QROWS) ? (x + (long)wid * KD)
                                   : (xn + (long)(wid - QROWS) * KD);
  float s = 0.0f;
  for (int i = lane; i < KD; i += 32) {
    float v = src[i];
    s = fmaf(v, v, s);
  }
  for (int off = 16; off; off >>= 1) s += __shfl_xor(s, off, 32);
  if (lane == 0) ws[wid] = s;
}

// ---------------------------------------------------------------------------
// Kernel 3: fused fp32-WMMA GEMM + streaming top-33 + scatter.
// One block per 16-query tile, 2 waves, each wave owns a 16x16 tile/chunk.
// B chunks are DMA'd by the Tensor Data Mover into double-buffered padded
// LDS (TENSORcnt pipelining), so global->LDS staging overlaps the WMMAs.
// ---------------------------------------------------------------------------
__global__ __launch_bounds__(64) void knn_kernel(const float* __restrict__ x,
                                                 const float* __restrict__ xn,
                                                 const float* __restrict__ x2g,
                                                 const float* __restrict__ y2g,
                                                 float* __restrict__ out) {
  __shared__ float As[16 * ASTRIDE];          // 16.6 KB  A tile (TDM-padded)
  __shared__ float Bs0[NCHUNK * BSTRIDE];     // 32.9 KB  B ping (TDM-padded)
  __shared__ float Bs1[NCHUNK * BSTRIDE];     // 32.9 KB  B pong
  __shared__ float d2buf[16 * NCHUNK];        //  2.0 KB
  __shared__ float topv[16 * TOPK];           //  2.1 KB  33 smallest d^2
  __shared__ int   topi[16 * TOPK];           //  2.1 KB
  __shared__ float x2s[16];
  __shared__ float worst[16];
  __shared__ int   wslot[16];
  __shared__ int   cnt[16];

  const int tid  = threadIdx.x;
  const int wave = tid >> 5;                  // 0..1
  const int lane = tid & 31;
  const int q0   = blockIdx.x * 16;

  const unsigned asBase = lds_off(As);
  const unsigned bBase[2] = {lds_off(Bs0), lds_off(Bs1)};

  // Prologue DMAs (each wave issues & tracks its own copy; identical bytes).
  // A tile: 16 x 256, pad 4 DW per 256 DW -> stride 260.
  tdm_load_2d(asBase, x + (long)q0 * KD, KD, 16, KD, QROWS, KD, 7, 3);
  // B chunk 0: 32 x 256, pad 1 DW per 256 DW -> stride 257.
  tdm_load_2d(bBase[0], xn, KD, NCHUNK, KD, NCOLS, KD, 7, 0);

  if (tid < 16) {
    cnt[tid]   = 0;
    worst[tid] = -3.4e38f;
    wslot[tid] = 0;
    x2s[tid]   = x2g[q0 + tid];
  }

  // Fragment geometry for V_WMMA_F32_16X16X4_F32 (32-bit A 16x4 layout:
  // lanes 0-15 hold K={0,1}, lanes 16-31 hold K={2,3}; B 4x16 analogous).
  const int mrow  = lane & 15;                // A row / B col within tile
  const int kh    = (lane >> 4) << 1;         // 0 or 2
  const int wtile = wave << 4;                // column tile base within chunk

  int cur = 0;
  for (int n0 = 0; n0 < NCOLS; n0 += NCHUNK) {
    // Pipeline: kick off chunk n0+NCHUNK into the other buffer, then wait for
    // the oldest outstanding DMA (this chunk). In-order TENSORcnt semantics.
    if (n0 + NCHUNK < NCOLS) {
      tdm_load_2d(bBase[cur ^ 1], xn + (long)(n0 + NCHUNK) * KD,
                  KD, NCHUNK, KD, NCOLS, KD, 7, 0);
      __builtin_amdgcn_s_wait_tensorcnt(1);
    } else {
      __builtin_amdgcn_s_wait_tensorcnt(0);
    }

    const float* Bcur = cur ? Bs1 : Bs0;
    const float* Ap = &As[mrow * ASTRIDE];
    const float* Bp = &Bcur[(wtile + mrow) * BSTRIDE];

    // 16x16 fp32 WMMA over K=256: 64 x v_wmma_f32_16x16x4_f32.
    v8f acc = {};
    for (int k0 = 0; k0 < KD; k0 += 4) {
      v2f a, b;
      a.x = Ap[k0 + kh];
      a.y = Ap[k0 + kh + 1];
      b.x = Bp[k0 + kh];
      b.y = Bp[k0 + kh + 1];
      acc = __builtin_amdgcn_wmma_f32_16x16x4_f32(
          /*neg_a=*/false, a, /*neg_b=*/false, b,
          /*c_mod=*/(short)0, acc, /*reuse_a=*/false, /*reuse_b=*/false);
    }

    // d^2 = x2 + y2 - 2*dot. C layout: lane<16 -> M=i, lane>=16 -> M=i+8.
    {
      float y2v   = y2g[n0 + wtile + mrow];
      int   mbase = (lane >> 4) << 3;
#pragma unroll
      for (int i = 0; i < 8; ++i) {
        int m = mbase + i;
        d2buf[m * NCHUNK + wtile + mrow] = x2s[m] + y2v - 2.0f * acc[i];
      }
    }
    __syncthreads();

    // Streaming selection: 16 lanes, one row each. Keep 33 smallest d^2.
    if (tid < 16) {
      const int r = tid;
      float w = worst[r];
      int   c = cnt[r];
      for (int j = 0; j < NCHUNK; ++j) {
        float d2 = d2buf[r * NCHUNK + j];
        if (c < TOPK) {
          topv[r * TOPK + c] = d2;
          topi[r * TOPK + c] = n0 + j;
          if (c == 0 || d2 > w) { w = d2; wslot[r] = c; }
          ++c;
        } else if (d2 < w) {
          int s = wslot[r];
          topv[r * TOPK + s] = d2;
          topi[r * TOPK + s] = n0 + j;
          w = topv[r * TOPK];
          s = 0;
          for (int jj = 1; jj < TOPK; ++jj) {
            float v = topv[r * TOPK + jj];
            if (v > w) { w = v; s = jj; }
          }
          wslot[r] = s;
        }
      }
      worst[r] = w;
      cnt[r]   = c;
    }
    __syncthreads();   // also fences buf[cur^1] reuse before next TDM issue
    cur ^= 1;
  }

  // Scatter the 33 winners per row: out = -sqrt(max(d2,0)); rest stays -inf.
  for (int t = tid; t < 16 * TOPK; t += 64) {
    int r = t / TOPK;
    int s = t - r * TOPK;
    if (s < cnt[r]) {
      float d2 = topv[r * TOPK + s];
      out[(long)(q0 + r) * NCOLS + topi[r * TOPK + s]] =
          -sqrtf(fmaxf(d2, 0.0f));
    }
  }
}

// ---------------------------------------------------------------------------
extern "C" void kernel_launch(void* const* d_in, const int* in_sizes, int n_in,
                              void* d_out, int out_size, void* d_ws,
                              size_t ws_size, hipStream_t stream) {
  (void)in_sizes; (void)n_in; (void)out_size; (void)ws_size;
  const float* x  = (const float*)d_in[0];
  const float* xn = (const float*)d_in[1];
  float* out = (float*)d_out;
  float* ws  = (float*)d_ws;   // [0,4096) x^2 ; [4096,20480) y^2  (80 KB)

  long n4 = (long)QROWS * NCOLS / 4;
  fill_neg_inf<<<16384, 256, 0, stream>>>((float4*)out, n4);

  int nrows = QROWS + NCOLS;                 // one wave per row
  int blocks = (nrows * 32 + 255) / 256;
  row_norms<<<blocks, 256, 0, stream>>>(x, xn, ws);

  knn_kernel<<<QROWS / 16, 64, 0, stream>>>(x, xn, ws, ws + QROWS, out);
}